// IPGR_43714177138865
// MI455X (gfx1250) — compile-verified
//
#include <hip/hip_runtime.h>
#include <hip/hip_bf16.h>

typedef float v2f __attribute__((ext_vector_type(2)));
typedef float v8f __attribute__((ext_vector_type(8)));

#define BATCH 4
#define NPTS  8192
#define MPTS  4096
#define TILES_M (MPTS / 16)            // 256
#define WAVES_PER_BLOCK 8
#define PRED_PER_BLOCK (16 * WAVES_PER_BLOCK)    // 128 pred points per block
#define BLOCKS_PER_BATCH (NPTS / PRED_PER_BLOCK) // 64

// ---------------------------------------------------------------------------
// Pack partial points into [x, y, z, |p|^2] float4 layout (loop-invariant).
// ---------------------------------------------------------------------------
__global__ void cdna5_pack_partial(const float* __restrict__ partial,
                                   float4* __restrict__ packed) {
    int i = blockIdx.x * blockDim.x + threadIdx.x;
    if (i >= BATCH * MPTS) return;
    const float* p = partial + (size_t)i * 3;
    float x = p[0], y = p[1], z = p[2];
    packed[i] = make_float4(x, y, z, x * x + y * y + z * z);
}

// ---------------------------------------------------------------------------
// Zero the per-batch max accumulators (float-as-uint, distances >= 0).
// ---------------------------------------------------------------------------
__global__ void cdna5_zero_max(unsigned int* __restrict__ batch_max) {
    if (threadIdx.x < BATCH) batch_max[threadIdx.x] = 0u;
}

// ---------------------------------------------------------------------------
// Nearest-neighbor kernel using V_WMMA_F32_16X16X4_F32.
//
// A (16x4, partial tile):  lane l<16 : (p.x, p.y)    lane l>=16 : (p.z, |p|^2)
// B (4x16, pred points):   lane l<16 : (-2x, -2y)    lane l>=16 : (-2z, 1.0)
// C[m][n] = |p_m|^2 - 2 p_m . x_n ;  d2 = C + |x_n|^2 (x_n^2 hoisted: it is a
// per-lane constant, so argmin over C == argmin over d2).
// C layout: VGPR r -> row m = r (lanes 0-15) / r+8 (lanes 16-31), col n = lane&15
//
// Min/argmin kept as 8 INDEPENDENT per-slot accumulators (one per C VGPR) so
// the compare/select ops have no cross-iteration serial chain and dual-issue
// alongside the XDL WMMA; per-slot index is just the tile counter t.
// ---------------------------------------------------------------------------
__global__ __launch_bounds__(256) void cdna5_nn_kernel(
        const float* __restrict__ refined,
        const float4* __restrict__ packed,
        float* __restrict__ min_dist,
        int* __restrict__ min_idx,
        unsigned int* __restrict__ batch_max) {

    __shared__ float4 lds_part[MPTS];   // 64 KB: packed partial for this batch

    const int b  = blockIdx.x / BLOCKS_PER_BATCH;
    const int tb = blockIdx.x % BLOCKS_PER_BATCH;

    // Stage this batch's packed partial into LDS (float4 copies).
    const float4* src = packed + (size_t)b * MPTS;
    for (int i = threadIdx.x; i < MPTS; i += blockDim.x) lds_part[i] = src[i];
    __syncthreads();

    const int wave = threadIdx.x >> 5;
    const int lane = threadIdx.x & 31;
    const int col  = lane & 15;
    const bool hi  = lane >= 16;
    const int rowOff = hi ? 8 : 0;

    // This lane's pred point (column n of the B matrix).
    const int n = tb * PRED_PER_BLOCK + wave * 16 + col;
    const float* rp = refined + ((size_t)b * NPTS + n) * 3;
    const float x = rp[0], y = rp[1], z = rp[2];
    const float x2 = x * x + y * y + z * z;

    v2f bfrag;
    if (!hi) { bfrag[0] = -2.0f * x; bfrag[1] = -2.0f * y; }
    else     { bfrag[0] = -2.0f * z; bfrag[1] = 1.0f;      }

    // Per-slot running minima (independent -> no serial VALU chain).
    float bestD[8];
    int   bestT[8];
#pragma unroll
    for (int r = 0; r < 8; ++r) { bestD[r] = 3.4e38f; bestT[r] = 0; }

    const float* ldsf = (const float*)lds_part;
    const int aoff = (hi ? 2 : 0) + col * 4;

#pragma unroll 4
    for (int t = 0; t < TILES_M; ++t) {
        // A fragment for this lane: partial point m = t*16 + col.
        const float* pp = ldsf + (size_t)t * 64 + aoff;
        v2f afrag;
        afrag[0] = pp[0];
        afrag[1] = pp[1];

        v8f c = {};
        c = __builtin_amdgcn_wmma_f32_16x16x4_f32(
                /*neg_a=*/false, afrag, /*neg_b=*/false, bfrag,
                /*c_mod=*/(short)0, c, /*reuse_a=*/false, /*reuse_b=*/false);

#pragma unroll
        for (int r = 0; r < 8; ++r) {
            bool lt = c[r] < bestD[r];
            bestD[r] = lt ? c[r] : bestD[r];
            bestT[r] = lt ? t    : bestT[r];
        }
    }

    // Merge the 8 slots (once per wave; lexicographic for deterministic ties).
    float best   = 3.4e38f;
    int   bestIdx = 0;
#pragma unroll
    for (int r = 0; r < 8; ++r) {
        float d2  = bestD[r] + x2;
        int   idx = bestT[r] * 16 + rowOff + r;
        if (d2 < best || (d2 == best && idx < bestIdx)) { best = d2; bestIdx = idx; }
    }

    // Combine the two lane-halves (rows 0-7 vs 8-15 of each tile).
    {
        float ob = __shfl_xor(best, 16, 32);
        int   oi = __shfl_xor(bestIdx, 16, 32);
        if (ob < best || (ob == best && oi < bestIdx)) { best = ob; bestIdx = oi; }
    }

    best = fmaxf(best, 0.0f);
    const float d = sqrtf(best + 1e-12f);

    if (!hi) {
        min_dist[(size_t)b * NPTS + n] = d;
        min_idx [(size_t)b * NPTS + n] = bestIdx;
    }

    // Wave-level max of min_dist (lanes j and j+16 hold identical values),
    // then one atomicMax per wave on float bits (valid: d >= 0).
    float m = d;
#pragma unroll
    for (int s = 8; s >= 1; s >>= 1) m = fmaxf(m, __shfl_xor(m, s, 32));
    if (lane == 0) atomicMax(batch_max + b, __float_as_uint(m));
}

// ---------------------------------------------------------------------------
// Pointwise refinement update.
// ---------------------------------------------------------------------------
__global__ void cdna5_update_kernel(float* __restrict__ refined,
                                    const float* __restrict__ partial,
                                    const float* __restrict__ min_dist,
                                    const int* __restrict__ min_idx,
                                    const unsigned int* __restrict__ batch_max) {
    int i = blockIdx.x * blockDim.x + threadIdx.x;
    if (i >= BATCH * NPTS) return;
    const int b = i / NPTS;

    const float maxv = __uint_as_float(batch_max[b]);
    const float md   = min_dist[i];
    const int   idx  = min_idx[i];

    const float* p = partial + ((size_t)b * MPTS + idx) * 3;
    const float alpha = 0.1f * (2.0f - md / (maxv + 1e-6f));

    float* r = refined + (size_t)i * 3;
    float rx = r[0], ry = r[1], rz = r[2];
    r[0] = rx + alpha * (p[0] - rx);
    r[1] = ry + alpha * (p[1] - ry);
    r[2] = rz + alpha * (p[2] - rz);
}

// ---------------------------------------------------------------------------
// Launcher: copy pred -> refined(d_out), pack partial once, then 4 iterations
// of {zero max, NN (WMMA), update}. All on `stream`, graph-capture safe.
// ---------------------------------------------------------------------------
extern "C" void kernel_launch(void* const* d_in, const int* in_sizes, int n_in,
                              void* d_out, int out_size, void* d_ws, size_t ws_size,
                              hipStream_t stream) {
    const float* pred    = (const float*)d_in[0];   // [4, 8192, 3]
    const float* partial = (const float*)d_in[1];   // [4, 4096, 3]
    float* refined = (float*)d_out;                 // [4, 8192, 3]

    // Workspace layout.
    char* ws = (char*)d_ws;
    float4*       packed    = (float4*)ws;                                   // 262144 B
    float*        min_dist  = (float*)(ws + (size_t)BATCH * MPTS * 16);      // 131072 B
    int*          min_idx   = (int*)(ws + (size_t)BATCH * MPTS * 16
                                        + (size_t)BATCH * NPTS * 4);         // 131072 B
    unsigned int* batch_max = (unsigned int*)(ws + (size_t)BATCH * MPTS * 16
                                                 + (size_t)BATCH * NPTS * 8);

    // refined = pred
    hipMemcpyAsync(refined, pred, (size_t)BATCH * NPTS * 3 * sizeof(float),
                   hipMemcpyDeviceToDevice, stream);

    // Pack partial once (loop-invariant).
    cdna5_pack_partial<<<(BATCH * MPTS + 255) / 256, 256, 0, stream>>>(partial, packed);

    for (int it = 0; it < 4; ++it) {
        cdna5_zero_max<<<1, 32, 0, stream>>>(batch_max);
        cdna5_nn_kernel<<<BATCH * BLOCKS_PER_BATCH, 256, 0, stream>>>(
            refined, packed, min_dist, min_idx, batch_max);
        cdna5_update_kernel<<<(BATCH * NPTS + 255) / 256, 256, 0, stream>>>(
            refined, partial, min_dist, min_idx, batch_max);
    }
}